// ASPPDeformable_52046413693303
// MI455X (gfx1250) — compile-verified
//
#include <hip/hip_runtime.h>
#include <hip/hip_bf16.h>
#include <math.h>

// ---------------------------------------------------------------------------
// ASPP-Deformable forward for MI455X (gfx1250).
// Heavy math via V_WMMA_F32_16X16X4_F32 (fp32 A/B, fp32 acc): full fp32
// precision on the CDNA5 matrix pipe. All GEMM staging is branch-free and
// vectorized (float4 global loads -> b128 LDS stores).
// ---------------------------------------------------------------------------

typedef __attribute__((ext_vector_type(2))) float v2f;
typedef __attribute__((ext_vector_type(8))) float v8f;

#define TM     32      // block tile M
#define TN     64      // block tile N
#define KSTEP  32      // K staged per iteration (all Kd are multiples of 32)
#define NP     4096    // spatial N = 64*64 (fixed)
#define ASTRIDE 36     // padded LDS strides (multiples of 4 -> 16B-aligned rows,
#define BSTRIDE 72     //  odd-ish mod 64 pattern -> conflict-free frag reads)

// ---------------------------------------------------------------------------
// fp32 WMMA GEMM: out[M, 4096] = A[M,Kd] * B[Kd,4096] (+bias) (+ACT)
//   ACT == 0: none ; ACT == 1: 2*sigmoid (modulator)
// 128 threads = 4 waves (2x2); each wave computes 16x32 (two 16x16 accs).
// Requires: Kd % 32 == 0 (true for 128/1152/1280/6272).
// ---------------------------------------------------------------------------
template<int ACT>
__global__ void __launch_bounds__(128)
gemm_f32_wmma(const float* __restrict__ A, const float* __restrict__ Bm,
              const float* __restrict__ bias, float* __restrict__ out,
              int M, int Kd)
{
    __shared__ float As[TM][ASTRIDE];      // [m][k]
    __shared__ float Bs[KSTEP][BSTRIDE];   // [k][n]

    const int tid   = threadIdx.x;
    const int wave  = tid >> 5;
    const int lane  = tid & 31;
    const int waveM = wave >> 1;           // 0..1
    const int waveN = wave & 1;            // 0..1
    const int l16   = lane & 15;
    const int half  = lane >> 4;           // 0: K{0,1}, 1: K{2,3}
    const int mBase = blockIdx.y * TM;
    const int nBase = blockIdx.x * TN;

    // staging coordinates (branch-free)
    const int aRow = tid >> 2;             // 0..31
    const int aK   = (tid & 3) * 8;        // 0,8,16,24
    const int bK   = tid >> 2;             // 0..31
    const int bN   = (tid & 3) * 16;       // 0,16,32,48

    const int  aRowG = mBase + aRow;
    const int  aRowC = aRowG < M ? aRowG : (M - 1);   // clamp: address always valid
    const bool aLive = aRowG < M;

    v8f acc0 = {0.f,0.f,0.f,0.f,0.f,0.f,0.f,0.f};
    v8f acc1 = {0.f,0.f,0.f,0.f,0.f,0.f,0.f,0.f};

    for (int k0 = 0; k0 < Kd; k0 += KSTEP) {
        // ---- stage A: 32x32, 8 floats/thread, contiguous along k ----
        {
            const float4* p = (const float4*)(A + (long)aRowC * Kd + k0 + aK);
            float4 v0 = p[0], v1 = p[1];
            if (!aLive) { v0 = make_float4(0.f,0.f,0.f,0.f); v1 = v0; }
            float4* d = (float4*)&As[aRow][aK];
            d[0] = v0; d[1] = v1;
        }
        // ---- stage B: 32x64, 16 floats/thread, contiguous along n ----
        {
            const float4* p = (const float4*)(Bm + (long)(k0 + bK) * NP + nBase + bN);
            float4 v0 = p[0], v1 = p[1], v2 = p[2], v3 = p[3];
            float4* d = (float4*)&Bs[bK][bN];
            d[0] = v0; d[1] = v1; d[2] = v2; d[3] = v3;
        }
        // prefetch next B panel into GL2 (global_prefetch_b8)
        if (k0 + KSTEP < Kd)
            __builtin_prefetch(Bm + (long)(k0 + KSTEP + bK) * NP + nBase + bN, 0, 0);
        __syncthreads();

        #pragma unroll
        for (int kc = 0; kc < KSTEP / 4; ++kc) {
            const int kb = kc * 4 + half * 2;
            v2f a, b0, b1;
            // A 16x4 fragment: lanes0-15 K={kb0,kb0+1}, lanes16-31 K={+2,+3}
            a.x = As[waveM * 16 + l16][kb];
            a.y = As[waveM * 16 + l16][kb + 1];
            // two B 4x16 fragments (N split 16+16 of this wave's 32 columns)
            b0.x = Bs[kb    ][waveN * 32 + l16];
            b0.y = Bs[kb + 1][waveN * 32 + l16];
            b1.x = Bs[kb    ][waveN * 32 + 16 + l16];
            b1.y = Bs[kb + 1][waveN * 32 + 16 + l16];
            acc0 = __builtin_amdgcn_wmma_f32_16x16x4_f32(
                false, a, false, b0, (short)0, acc0, false, false);
            acc1 = __builtin_amdgcn_wmma_f32_16x16x4_f32(
                false, a, false, b1, (short)0, acc1, false, false);
        }
        __syncthreads();
    }

    // C/D layout: VGPR v -> M = v + 8*half, N = l16
    const int colg0 = nBase + waveN * 32 + l16;
    #pragma unroll
    for (int v = 0; v < 8; ++v) {
        int rowg = mBase + waveM * 16 + half * 8 + v;
        if (rowg < M) {
            float bv = bias ? bias[rowg] : 0.f;
            float x0 = acc0[v] + bv;
            float x1 = acc1[v] + bv;
            if (ACT == 1) {
                x0 = 2.f / (1.f + __expf(-x0));
                x1 = 2.f / (1.f + __expf(-x1));
            }
            out[(long)rowg * NP + colg0]      = x0;
            out[(long)rowg * NP + colg0 + 16] = x1;
        }
    }
}

// ---------------------------------------------------------------------------
// Plain im2col (compile-time K/PAD; C=128, H=W=64 fixed):
//   cols[(c*K*K + ky*K + kx) * 4096 + n]
// ---------------------------------------------------------------------------
template<int K, int PAD>
__global__ void __launch_bounds__(256)
im2col_t(const float* __restrict__ xb, float* __restrict__ cols)
{
    constexpr int KK = K * K;
    const long total = 128L * KK * NP;
    long i = (long)blockIdx.x * 256 + threadIdx.x;
    if (i >= total) return;
    int n  = (int)(i & (NP - 1));
    int kg = (int)(i >> 12);
    int c  = kg / KK;
    int kk = kg % KK;
    int ky = kk / K, kx = kk % K;
    int y  = n >> 6, xq = n & 63;
    int iy = y - PAD + ky, ix = xq - PAD + kx;
    float v = 0.f;
    if (iy >= 0 && iy < 64 && ix >= 0 && ix < 64)
        v = xb[((long)c << 12) + (iy << 6) + ix];
    cols[i] = v;
}

// ---------------------------------------------------------------------------
// Deformable im2col: bilinear (zero outside, torchvision) * mask.
// off channel 2*kk = dy, 2*kk+1 = dx; mask already 2*sigmoid().
// ---------------------------------------------------------------------------
template<int K, int PAD>
__global__ void __launch_bounds__(256)
deform_im2col_t(const float* __restrict__ xb, const float* __restrict__ off,
                const float* __restrict__ mask, float* __restrict__ cols)
{
    constexpr int KK = K * K;
    const long total = 128L * KK * NP;
    long i = (long)blockIdx.x * 256 + threadIdx.x;
    if (i >= total) return;
    int n  = (int)(i & (NP - 1));
    int kg = (int)(i >> 12);
    int c  = kg / KK;
    int kk = kg % KK;
    int ky = kk / K, kx = kk % K;
    int y  = n >> 6, xq = n & 63;

    float dy = off[((long)(2 * kk)     << 12) + n];
    float dx = off[((long)(2 * kk + 1) << 12) + n];
    float mv = mask[((long)kk << 12) + n];

    float py = (float)(y - PAD + ky) + dy;
    float px = (float)(xq - PAD + kx) + dx;
    float y0 = floorf(py), x0 = floorf(px);
    float wy = py - y0,    wx = px - x0;
    int iy0 = (int)y0, ix0 = (int)x0;

    const float* xc = xb + ((long)c << 12);
    auto samp = [&](int yy, int xx) -> float {
        return (yy >= 0 && yy < 64 && xx >= 0 && xx < 64) ? xc[(yy << 6) + xx] : 0.f;
    };
    float v00 = samp(iy0,     ix0);
    float v01 = samp(iy0,     ix0 + 1);
    float v10 = samp(iy0 + 1, ix0);
    float v11 = samp(iy0 + 1, ix0 + 1);
    float val = v00 * (1.f - wy) * (1.f - wx) + v01 * (1.f - wy) * wx
              + v10 * wy * (1.f - wx)         + v11 * wy * wx;
    cols[i] = val * mv;
}

// ---------------------------------------------------------------------------
// In-place GroupNorm + ReLU over a [Ctot, 4096] slice (one block per group).
// ---------------------------------------------------------------------------
__global__ void __launch_bounds__(256)
gn_relu_k(float* __restrict__ data, const float* __restrict__ gamma,
          const float* __restrict__ beta, int cpg)
{
    __shared__ float ssum[256], ssq[256];
    const int g = blockIdx.x, tid = threadIdx.x;
    const long base = (long)g * cpg * NP;
    const long cnt  = (long)cpg * NP;
    float s = 0.f, sq = 0.f;
    for (long i = tid; i < cnt; i += 256) {
        float v = data[base + i];
        s += v; sq += v * v;
    }
    ssum[tid] = s; ssq[tid] = sq;
    __syncthreads();
    for (int o = 128; o > 0; o >>= 1) {
        if (tid < o) { ssum[tid] += ssum[tid + o]; ssq[tid] += ssq[tid + o]; }
        __syncthreads();
    }
    float mean = ssum[0] / (float)cnt;
    float var  = ssq[0] / (float)cnt - mean * mean;
    var = var < 0.f ? 0.f : var;
    float rstd = rsqrtf(var + 1e-5f);
    for (long i = tid; i < cnt; i += 256) {
        int ch = g * cpg + (int)(i >> 12);
        float v = (data[base + i] - mean) * rstd * gamma[ch] + beta[ch];
        data[base + i] = v > 0.f ? v : 0.f;
    }
}

// ---------------------------------------------------------------------------
// Spatial mean: xm[b*C+c] = mean_n x[(b*C+c)*4096 + n]
// ---------------------------------------------------------------------------
__global__ void __launch_bounds__(256)
mean_pool_k(const float* __restrict__ x, float* __restrict__ xm)
{
    __shared__ float ss[256];
    const int bc = blockIdx.x, tid = threadIdx.x;
    const float* p = x + ((long)bc << 12);
    float s = 0.f;
    for (int i = tid; i < NP; i += 256) s += p[i];
    ss[tid] = s;
    __syncthreads();
    for (int o = 128; o > 0; o >>= 1) {
        if (tid < o) ss[tid] += ss[tid + o];
        __syncthreads();
    }
    if (tid == 0) xm[bc] = ss[0] * (1.f / (float)NP);
}

// ---------------------------------------------------------------------------
// Pool branch: 1x1 conv on pooled vector -> GN(16 groups, spatial 1) -> ReLU
// -> broadcast into cat channels [1024,1280). grid.x = B, 256 threads.
// ---------------------------------------------------------------------------
__global__ void __launch_bounds__(256)
pool_branch_k(const float* __restrict__ xm, const float* __restrict__ pw,
              const float* __restrict__ gng, const float* __restrict__ gnb,
              float* __restrict__ cat)
{
    __shared__ float vals[256], outs[256];
    const int b = blockIdx.x, o = threadIdx.x;
    const float* xmb = xm + (long)b * 128;
    float acc = 0.f;
    for (int c = 0; c < 128; ++c) acc += pw[(long)o * 128 + c] * xmb[c];
    vals[o] = acc;
    __syncthreads();
    const int g = o >> 4;
    float m = 0.f;
    for (int j = 0; j < 16; ++j) m += vals[g * 16 + j];
    m *= (1.f / 16.f);
    float vv = 0.f;
    for (int j = 0; j < 16; ++j) { float d = vals[g * 16 + j] - m; vv += d * d; }
    vv *= (1.f / 16.f);
    float nv = (acc - m) * rsqrtf(vv + 1e-5f) * gng[o] + gnb[o];
    outs[o] = nv > 0.f ? nv : 0.f;
    __syncthreads();
    float* dst = cat + ((long)b * 1280 + 1024) * NP;
    for (long i = o; i < 256L * NP; i += 256)
        dst[i] = outs[(int)(i >> 12)];
}

// ---------------------------------------------------------------------------
extern "C" void kernel_launch(void* const* d_in, const int* in_sizes, int n_in,
                              void* d_out, int out_size, void* d_ws, size_t ws_size,
                              hipStream_t stream)
{
    (void)in_sizes; (void)n_in; (void)out_size; (void)ws_size;

    const int Bz = 2, C = 128, O = 256;
    const int ks[4] = {1, 1, 3, 7};

    const float* x = (const float*)d_in[0];
    struct Br { const float *offw, *offb, *modw, *modb, *regw, *gng, *gnb; };
    Br br[4];
    int idx = 1;
    for (int i = 0; i < 4; ++i) {
        br[i].offw = (const float*)d_in[idx++];
        br[i].offb = (const float*)d_in[idx++];
        br[i].modw = (const float*)d_in[idx++];
        br[i].modb = (const float*)d_in[idx++];
        br[i].regw = (const float*)d_in[idx++];
        br[i].gng  = (const float*)d_in[idx++];
        br[i].gnb  = (const float*)d_in[idx++];
    }
    const float* pool_w   = (const float*)d_in[idx++];
    const float* pool_gng = (const float*)d_in[idx++];
    const float* pool_gnb = (const float*)d_in[idx++];
    const float* fuse_w   = (const float*)d_in[idx++];
    const float* fuse_gng = (const float*)d_in[idx++];
    const float* fuse_gnb = (const float*)d_in[idx++];

    // workspace layout (floats)
    float* ws       = (float*)d_ws;
    float* cols     = ws;                       // max 6272*4096 = 25,690,112
    float* off_buf  = cols    + 25690112L;      // 98*4096
    float* mask_buf = off_buf + 401408L;        // 49*4096
    float* cat      = mask_buf + 200704L;       // 2*1280*4096
    float* xm       = cat     + 10485760L;      // 2*128

    const dim3 blk(128);

    for (int b = 0; b < Bz; ++b) {
        const float* xb = x + ((long)b * C << 12);
        for (int bi = 0; bi < 4; ++bi) {
            const int k = ks[bi];
            const int KK = k * k, Kd = C * KK;
            const long tot = (long)Kd * NP;
            const unsigned nblk = (unsigned)((tot + 255) / 256);

            const float* colsB;
            if (k == 1) {
                colsB = xb;                     // 1x1/pad0: im2col == x
            } else if (k == 3) {
                im2col_t<3, 1><<<nblk, 256, 0, stream>>>(xb, cols);
                colsB = cols;
            } else {
                im2col_t<7, 3><<<nblk, 256, 0, stream>>>(xb, cols);
                colsB = cols;
            }
            // offset conv (M = 2k^2) and modulator conv + 2*sigmoid (M = k^2)
            {
                dim3 g(NP / TN, (2 * KK + TM - 1) / TM);
                gemm_f32_wmma<0><<<g, blk, 0, stream>>>(
                    br[bi].offw, colsB, br[bi].offb, off_buf, 2 * KK, Kd);
            }
            {
                dim3 g(NP / TN, (KK + TM - 1) / TM);
                gemm_f32_wmma<1><<<g, blk, 0, stream>>>(
                    br[bi].modw, colsB, br[bi].modb, mask_buf, KK, Kd);
            }
            // bilinear-sampled, masked columns
            if (k == 1)      deform_im2col_t<1, 0><<<nblk, 256, 0, stream>>>(xb, off_buf, mask_buf, cols);
            else if (k == 3) deform_im2col_t<3, 1><<<nblk, 256, 0, stream>>>(xb, off_buf, mask_buf, cols);
            else             deform_im2col_t<7, 3><<<nblk, 256, 0, stream>>>(xb, off_buf, mask_buf, cols);

            // aggregation GEMM into concat slice, then GN+ReLU in place
            float* slice = cat + (((long)b * 1280 + (long)bi * 256) << 12);
            {
                dim3 g(NP / TN, O / TM);
                gemm_f32_wmma<0><<<g, blk, 0, stream>>>(
                    br[bi].regw, cols, nullptr, slice, O, Kd);
            }
            gn_relu_k<<<16, 256, 0, stream>>>(slice, br[bi].gng, br[bi].gnb, 16);
        }
    }

    // pool branch
    mean_pool_k<<<Bz * C, 256, 0, stream>>>(x, xm);
    pool_branch_k<<<Bz, 256, 0, stream>>>(xm, pool_w, pool_gng, pool_gnb, cat);

    // fuse 1x1 conv (GEMM directly on concat) + GN + ReLU
    for (int b = 0; b < Bz; ++b) {
        float* ob = (float*)d_out + ((long)b * C << 12);
        dim3 g(NP / TN, C / TM);
        gemm_f32_wmma<0><<<g, blk, 0, stream>>>(
            fuse_w, cat + ((long)b * 1280 << 12), nullptr, ob, C, 1280);
        gn_relu_k<<<16, 256, 0, stream>>>(ob, fuse_gng, fuse_gnb, 8);
    }
}